// Main_vertical_68221260530287
// MI455X (gfx1250) — compile-verified
//
#include <hip/hip_runtime.h>

typedef __attribute__((ext_vector_type(2))) float v2f;
typedef __attribute__((ext_vector_type(8))) float v8f;

// module uses pi = 3.14159 (not math pi)
#define INV_SQRT_2PI   0.39894244f
#define NEG_HALF_LOG2E -0.72134752f   // -0.5 * log2(e)

// One wave32 handles a 16-row tile. Masked z^2 values are loaded directly in
// V_WMMA A-operand layout (lane m = row m, VGPR pair = K 2*half+{0,1}) and
// reduced with D = A x ones via v_wmma_f32_16x16x4_f32 (4 chained K-chunks).
__global__ __launch_bounds__(256) void prod_density_wmma_kernel(
    const float* __restrict__ values,
    const int*   __restrict__ lengths,
    const float* __restrict__ thetas,
    float*       __restrict__ out,
    int B)
{
    __shared__ float s_sum[8 * 16];          // 8 waves/block, 16 row-sums each

    const int lane = threadIdx.x & 31;
    const int wid  = threadIdx.x >> 5;
    const int m    = lane & 15;              // A-matrix row within tile
    const int half = lane >> 4;              // K-half select (A layout)

    const long r0   = ((long)blockIdx.x * 8 + wid) * 16;   // first row of tile
    long row        = r0 + m;
    if (row > (long)B - 1) row = (long)B - 1;              // clamp, keep EXEC full

    // Uniform learned parameters (broadcast through L2/K$).
    const float t0    = thetas[0];
    const float scale = thetas[1];
    const float loc   = thetas[2];
    const float l5    = fmaxf(fminf(t0, 1.0f), 0.0f);
    const float l1    = 1.0f - l5;           // continue-prob complement
    const float inv_s = 1.0f / scale;
    const float Cc    = l5 * INV_SQRT_2PI * inv_s;   // per-token coefficient
    const float log2C = __log2f(Cc);

    const int n = lengths[row];

    // Accumulate D[m][*] = sum_j masked_z2[m][j] with B = ones (row-sum trick).
    v8f acc = {};
    v2f ones; ones.x = 1.0f; ones.y = 1.0f;

    const float* rowp = values + row * 16 + half * 2;

    #pragma unroll
    for (int c = 0; c < 4; ++c) {
        v2f v = *(const v2f*)(rowp + c * 4);         // global_load_b64, A layout
        const int col0 = c * 4 + half * 2;
        float z0 = (v.x - loc) * inv_s;
        float z1 = (v.y - loc) * inv_s;
        v2f a;
        a.x = (col0     < n) ? z0 * z0 : 0.0f;       // mask folds into A
        a.y = (col0 + 1 < n) ? z1 * z1 : 0.0f;
        acc = __builtin_amdgcn_wmma_f32_16x16x4_f32(
                  /*neg_a=*/false, a, /*neg_b=*/false, ones,
                  /*c_mod=*/(short)0, acc, /*reuse_a=*/false, /*reuse_b=*/false);
    }

    // D layout: lane<16 holds rows 0..7 in acc[0..7]; lane>=16 holds rows 8..15.
    // Bounce 16 sums per wave through LDS so lane m ends up owning row m.
    if (m == 0) {
        const int base = wid * 16 + half * 8;
        #pragma unroll
        for (int v = 0; v < 8; ++v) s_sum[base + v] = acc[v];
    }
    __syncthreads();

    if (lane < 16) {
        const long r = r0 + lane;
        if (r < (long)B) {
            const float S = s_sum[wid * 16 + lane];           // sum of masked z^2
            const float e = (n == 0) ? 0.0f : (float)n * log2C;
            out[r] = l1 * exp2f(e + NEG_HALF_LOG2E * S);      // single v_exp_f32
        }
    }
}

extern "C" void kernel_launch(void* const* d_in, const int* in_sizes, int n_in,
                              void* d_out, int out_size, void* d_ws, size_t ws_size,
                              hipStream_t stream) {
    const float* values  = (const float*)d_in[0];
    const int*   lengths = (const int*)d_in[1];
    const float* thetas  = (const float*)d_in[2];
    float*       out     = (float*)d_out;

    const int B = in_sizes[1];               // lengths has one entry per row
    const int rows_per_block = 128;          // 8 waves x 16 rows
    const int grid = (B + rows_per_block - 1) / rows_per_block;

    prod_density_wmma_kernel<<<grid, 256, 0, stream>>>(values, lengths, thetas, out, B);
}